// XEdgeConv_12584254178059
// MI455X (gfx1250) — compile-verified
//
#include <hip/hip_runtime.h>
#include <math.h>

typedef __attribute__((ext_vector_type(2))) float v2f;
typedef __attribute__((ext_vector_type(4))) float v4f;
typedef __attribute__((ext_vector_type(8))) float v8f;

#define BB 4
#define DD 64
#define NN 16384
#define KK 20

// Build W' (128x64) = [ Wa ; Wb - Wa ] from W (64x128) = [Wa | Wb].
// Row o of y-part projects neighbor features; row o of c-part projects the center.
__global__ __launch_bounds__(256) void prep_w_kernel(const float* __restrict__ W1,
                                                     const float* __restrict__ W2,
                                                     float* __restrict__ Wp1,
                                                     float* __restrict__ Wp2) {
    int i = blockIdx.x * blockDim.x + threadIdx.x;   // exactly 128*64 threads
    int m = i >> 6, k = i & 63;
    if (m < 64) {
        Wp1[i] = W1[m * 128 + k];
        Wp2[i] = W2[m * 128 + k];
    } else {
        int mm = m - 64;
        Wp1[i] = W1[mm * 128 + 64 + k] - W1[mm * 128 + k];
        Wp2[i] = W2[mm * 128 + 64 + k] - W2[mm * 128 + k];
    }
}

// z = W'(128x64) @ X(64xN) per batch, via V_WMMA_F32_16X16X4_F32.
// One wave owns one 16-column tile and all 8 row tiles (8 x v8f accumulators).
// Output stored transposed: ZT[b][n][0..127]  ([y | c]) so gathers are coalesced.
// LAYOUT 0: X is (B, 64, N) channel-major (original input x).
// LAYOUT 1: X is (B, N, 64) point-major (hidden activations h).
template <int LAYOUT>
__global__ __launch_bounds__(256) void gemm_zt_kernel(const float* __restrict__ Wp,
                                                      const float* __restrict__ X,
                                                      float* __restrict__ ZT) {
    const int lane = threadIdx.x & 31;
    const int wave = (blockIdx.x * blockDim.x + threadIdx.x) >> 5;
    const int tilesPerBatch = NN / 16;
    const int b = wave / tilesPerBatch;
    const int t = wave - b * tilesPerBatch;
    const int colLocal = lane & 15;       // N within tile (and M within 16 for A)
    const int col = t * 16 + colLocal;    // global point index
    const int khalf = lane >> 4;          // K sub-slot: 0 -> K{0,1}, 1 -> K{2,3}

    v8f acc[8] = {};

    for (int k0 = 0; k0 < 64; k0 += 4) {
        const int kk = k0 + khalf * 2;
        v2f bm;
        if (LAYOUT == 0) {
            const float* xb = X + (size_t)b * DD * NN;
            bm.x = xb[(size_t)kk * NN + col];
            bm.y = xb[(size_t)(kk + 1) * NN + col];
        } else {
            const float* xb = X + ((size_t)b * NN + col) * DD + kk;
            bm.x = xb[0];
            bm.y = xb[1];
        }
#pragma unroll
        for (int mt = 0; mt < 8; ++mt) {
            const float* wp = Wp + (mt * 16 + colLocal) * 64 + kk;
            v2f am;
            am.x = wp[0];
            am.y = wp[1];
            acc[mt] = __builtin_amdgcn_wmma_f32_16x16x4_f32(
                false, am, false, bm, (short)0, acc[mt], false, false);
        }
    }

    // Accumulator: VGPR r holds M = mt*16 + khalf*8 + r for column `col`.
    // Lane writes 8 contiguous channels per tile -> two b128 stores (transposed store).
    float* outp = ZT + ((size_t)b * NN + col) * 128 + khalf * 8;
#pragma unroll
    for (int mt = 0; mt < 8; ++mt) {
        v4f lo = {acc[mt][0], acc[mt][1], acc[mt][2], acc[mt][3]};
        v4f hi = {acc[mt][4], acc[mt][5], acc[mt][6], acc[mt][7]};
        *(v4f*)(outp + mt * 16) = lo;
        *(v4f*)(outp + mt * 16 + 4) = hi;
    }
}

__device__ __forceinline__ float gelu_exact(float x) {
    return 0.5f * x * (1.0f + erff(x * 0.70710678118654752f));
}

// Block 1 tail: wave-per-point; lane owns channel pair (2l, 2l+1).
// m[o] = max_k y[idx_k][o];  h = gelu(bn1(m + c[n]))  stored (B, N, 64).
__global__ __launch_bounds__(256) void gather1_kernel(const float* __restrict__ ZT,
                                                      const int* __restrict__ IDX,
                                                      const float* __restrict__ g1,
                                                      const float* __restrict__ b1,
                                                      const float* __restrict__ m1,
                                                      const float* __restrict__ v1,
                                                      float* __restrict__ H) {
    const int lane = threadIdx.x & 31;
    const int wave = (blockIdx.x * blockDim.x + threadIdx.x) >> 5;
    const int b = wave / NN;
    const int n = wave - b * NN;
    const int ch = lane * 2;
    const int* ip = IDX + ((size_t)b * NN + n) * KK;
    const float* zb = ZT + (size_t)b * NN * 128;

    float mx0 = -3.402823466e38f, mx1 = -3.402823466e38f;
    for (int k = 0; k < KK; ++k) {
        int j = ip[k];                           // uniform across wave -> broadcast
        const float* p = zb + (size_t)j * 128 + ch;  // 256B coalesced per wave
        mx0 = fmaxf(mx0, p[0]);
        mx1 = fmaxf(mx1, p[1]);
    }
    const float* cp = zb + (size_t)n * 128 + 64 + ch;
    float h0 = mx0 + cp[0];
    float h1 = mx1 + cp[1];

    float sc0 = g1[ch] / sqrtf(v1[ch] + 1e-5f);
    float sc1 = g1[ch + 1] / sqrtf(v1[ch + 1] + 1e-5f);
    float sh0 = b1[ch] - m1[ch] * sc0;
    float sh1 = b1[ch + 1] - m1[ch + 1] * sc1;

    float* hp = H + ((size_t)b * NN + n) * 64 + ch;
    hp[0] = gelu_exact(h0 * sc0 + sh0);
    hp[1] = gelu_exact(h1 * sc1 + sh1);
}

// Block 2 tail with identity shortcut: out = gelu(bn2(x + max_k y2 + c2)), (B, 64, N).
__global__ __launch_bounds__(256) void gather2_kernel(const float* __restrict__ ZT,
                                                      const int* __restrict__ IDX,
                                                      const float* __restrict__ X,
                                                      const float* __restrict__ g2,
                                                      const float* __restrict__ b2,
                                                      const float* __restrict__ m2,
                                                      const float* __restrict__ v2,
                                                      float* __restrict__ OUT) {
    const int lane = threadIdx.x & 31;
    const int wave = (blockIdx.x * blockDim.x + threadIdx.x) >> 5;
    const int b = wave / NN;
    const int n = wave - b * NN;
    const int ch = lane * 2;
    const int* ip = IDX + ((size_t)b * NN + n) * KK;
    const float* zb = ZT + (size_t)b * NN * 128;

    float mx0 = -3.402823466e38f, mx1 = -3.402823466e38f;
    for (int k = 0; k < KK; ++k) {
        int j = ip[k];
        const float* p = zb + (size_t)j * 128 + ch;
        mx0 = fmaxf(mx0, p[0]);
        mx1 = fmaxf(mx1, p[1]);
    }
    const float* cp = zb + (size_t)n * 128 + 64 + ch;
    float x0 = X[((size_t)b * DD + ch) * NN + n];
    float x1 = X[((size_t)b * DD + ch + 1) * NN + n];
    float h0 = x0 + mx0 + cp[0];
    float h1 = x1 + mx1 + cp[1];

    float sc0 = g2[ch] / sqrtf(v2[ch] + 1e-5f);
    float sc1 = g2[ch + 1] / sqrtf(v2[ch + 1] + 1e-5f);
    float sh0 = b2[ch] - m2[ch] * sc0;
    float sh1 = b2[ch + 1] - m2[ch + 1] * sc1;

    OUT[((size_t)b * DD + ch) * NN + n] = gelu_exact(h0 * sc0 + sh0);
    OUT[((size_t)b * DD + ch + 1) * NN + n] = gelu_exact(h1 * sc1 + sh1);
}

extern "C" void kernel_launch(void* const* d_in, const int* in_sizes, int n_in,
                              void* d_out, int out_size, void* d_ws, size_t ws_size,
                              hipStream_t stream) {
    const float* x   = (const float*)d_in[0];   // (B, 64, N)
    const int*   idx = (const int*)d_in[1];     // (B, N, 20)
    const float* W1  = (const float*)d_in[2];   // (64, 128)
    const float* W2  = (const float*)d_in[3];   // (64, 128)
    const float* g1  = (const float*)d_in[4];
    const float* b1  = (const float*)d_in[5];
    const float* m1  = (const float*)d_in[6];
    const float* v1  = (const float*)d_in[7];
    const float* g2  = (const float*)d_in[8];
    const float* b2  = (const float*)d_in[9];
    const float* m2  = (const float*)d_in[10];
    const float* v2  = (const float*)d_in[11];
    float* out = (float*)d_out;

    char* ws = (char*)d_ws;
    float* Wp1 = (float*)ws;                                   // 32 KB
    float* Wp2 = (float*)(ws + 32 * 1024);                     // 32 KB
    float* ZT  = (float*)(ws + 64 * 1024);                     // B*N*128*4 = 32 MiB
    float* H   = (float*)(ws + 64 * 1024 + (size_t)BB * NN * 128 * 4);  // B*N*64*4

    // W' construction (8192 threads exactly)
    prep_w_kernel<<<(128 * 64) / 256, 256, 0, stream>>>(W1, W2, Wp1, Wp2);

    const int gemmBlocks = (BB * (NN / 16) * 32) / 256;   // 4096 waves -> 512 blocks
    const int gathBlocks = (BB * NN * 32) / 256;          // 65536 waves -> 8192 blocks

    // Block 1
    gemm_zt_kernel<0><<<gemmBlocks, 256, 0, stream>>>(Wp1, x, ZT);
    gather1_kernel<<<gathBlocks, 256, 0, stream>>>(ZT, idx, g1, b1, m1, v1, H);
    // Block 2 (ZT reused; prior consumer finished in stream order)
    gemm_zt_kernel<1><<<gemmBlocks, 256, 0, stream>>>(Wp2, H, ZT);
    gather2_kernel<<<gathBlocks, 256, 0, stream>>>(ZT, idx, x, g2, b2, m2, v2, out);
}